// NonlinearFilter_89060441850597
// MI455X (gfx1250) — compile-verified
//
#include <hip/hip_runtime.h>
#include <hip/hip_bf16.h>
#include <math.h>

// Problem constants (from reference): B,T,L,S,H = 64,200,64,25,128
#define Bd 64
#define Td 200
#define Ld 64
#define Sd 25
#define Hd 128
#define SP 32   // S padded to 32 rows for 16x16 WMMA tiling

typedef __attribute__((ext_vector_type(2))) float v2f;
typedef __attribute__((ext_vector_type(8))) float v8f;

// Output offsets (flat f32 elements, in return order)
#define OFF_ZF   0
#define OFF_MF   (Sd*Bd*Td*Ld)                 // 20,480,000
#define OFF_MP   (OFF_MF + Bd*Td*Ld)
#define OFF_PF   (OFF_MP + Bd*Td*Ld)
#define OFF_CH   (OFF_PF + Bd*Td*Ld)

__device__ __forceinline__ float softplusf(float x) {
    // numerically stable log(1+exp(x))
    return fmaxf(x, 0.0f) + log1pf(expf(-fabsf(x)));
}

// Branch-free tanh on the hardware transcendental pipe (v_exp_f32).
// tanh(x) = 1 - 2/(exp(2x)+1); exp(+inf)=inf -> 1, exp(-inf)=0 -> -1.
__device__ __forceinline__ float fast_tanhf(float x) {
    float e = __expf(2.0f * x);
    return 1.0f - 2.0f / (e + 1.0f);
}

// D = A(16x4) * B(4x16) + C  in f32 on the matrix pipe.
__device__ __forceinline__ v8f wmma_f32_4(v2f a, v2f b, v8f c) {
    return __builtin_amdgcn_wmma_f32_16x16x4_f32(false, a, false, b, (short)0, c,
                                                 false, false);
}

__global__ __launch_bounds__(256, 1)
void nlf_kernel(const float* __restrict__ kin, const float* __restrict__ Kin,
                const float* __restrict__ win, const float* __restrict__ m0,
                const float* __restrict__ logQ0, const float* __restrict__ logQ,
                const float* __restrict__ W1, const float* __restrict__ b1,
                const float* __restrict__ W2, const float* __restrict__ b2,
                float* __restrict__ out) {
    __shared__ float W1s[Ld * Hd];    // 64x128
    __shared__ float W2s[Hd * Ld];    // 128x64
    __shared__ float b1s[Hd];
    __shared__ float b2s[Ld];
    __shared__ float Qd[Ld];          // softplus(log_Q)
    __shared__ float Zs[SP * Ld];     // z samples, rows 25..31 zero
    __shared__ float Hs[SP * Hd];     // tanh(z@W1+b1)
    __shared__ float Ms[SP * Ld];     // m = mean_fn(z), pad rows zero
    __shared__ float Ps[Ld * Ld];     // P_p, then Cholesky in-place
    __shared__ float mpS[Ld];
    __shared__ float mfS[Ld];
    __shared__ float sPfS[Ld];

    const int b   = blockIdx.x;       // trial index
    const int tid = threadIdx.x;      // 0..255
    const int wv  = tid >> 5;         // wave id 0..7
    const int lid = tid & 31;         // lane id
    const int ln  = lid & 15;         // lane mod 16
    const int khalf = (lid >> 4) * 2; // lane-half -> K offset {0,2}
    const int mhalf = (lid >> 4) * 8; // lane-half -> M offset in C/D
    const int ty  = tid >> 4;         // 0..15 (Cholesky 2D update)
    const int tx  = tid & 15;         // 0..15

    float* zf  = out + OFF_ZF;
    float* mfO = out + OFF_MF;
    float* mpO = out + OFF_MP;
    float* pfO = out + OFF_PF;
    float* chO = out + OFF_CH;

    // ---- Preload weights / constants into LDS --------------------------
    for (int p = tid; p < Ld * Hd; p += 256) W1s[p] = W1[p];
    for (int p = tid; p < Hd * Ld; p += 256) W2s[p] = W2[p];
    if (tid < Hd) b1s[tid] = b1[tid];
    if (tid < Ld) { b2s[tid] = b2[tid]; Qd[tid] = softplusf(logQ[tid]); }
    for (int p = tid; p < SP * Ld; p += 256) Zs[p] = 0.0f;  // zero pad rows
    __syncthreads();

    // ---- t = 0 (closed-form init) --------------------------------------
    if (tid < Ld) {
        const int l = tid;
        float P0  = softplusf(logQ0[l]);
        float J0  = 1.0f / P0;
        float h0  = J0 * m0[l];
        float Jf0 = J0 + Kin[(b * Td) * Ld + l];
        float Pf0 = 1.0f / Jf0;
        float mf0 = Pf0 * (h0 + kin[(b * Td) * Ld + l]);
        mfS[l]  = mf0;
        sPfS[l] = sqrtf(Pf0);
        mpS[l]  = sqrtf(P0);          // reuse mpS for sqrt(P0) at t=0
        mfO[(b * Td) * Ld + l] = mf0;
        mpO[(b * Td) * Ld + l] = m0[l];
        pfO[(b * Td) * Ld + l] = Pf0;
    }
    __syncthreads();
    for (int p = tid; p < Sd * Ld; p += 256) {
        int s = p >> 6, l = p & 63;
        float zv = mfS[l] + sPfS[l] * win[(s * Bd + b) * Ld + l];  // w[0,s,b,l]
        Zs[s * Ld + l] = zv;
        zf[s * (Bd * Td * Ld) + b * (Td * Ld) + l] = zv;
    }
    for (int p = tid; p < Ld * Ld; p += 256) {
        int i = p >> 6, j = p & 63;
        chO[((b * Td) * Ld + i) * Ld + j] = (i == j) ? mpS[i] : 0.0f;
    }

    // ---- main scan -----------------------------------------------------
    for (int t = 1; t < Td; ++t) {
        __syncthreads();   // Zs ready for this step

        // (1) Hpre = Zs(32x64) @ W1(64x128); Hs = tanh(Hpre + b1)
        //     16 tiles (2x8), 2 per wave, K-loop 64/4
        for (int q = 0; q < 2; ++q) {
            int tile = wv * 2 + q;
            int mi = tile >> 3, ni = tile & 7;
            int row = mi * 16 + ln;
            int col = ni * 16 + ln;
            v8f c = {};
            #pragma unroll
            for (int kk = 0; kk < 16; ++kk) {
                int kb = kk * 4 + khalf;
                v2f a, bb;
                a.x  = Zs[row * Ld + kb];       a.y  = Zs[row * Ld + kb + 1];
                bb.x = W1s[kb * Hd + col];      bb.y = W1s[(kb + 1) * Hd + col];
                c = wmma_f32_4(a, bb, c);
            }
            int mb = mi * 16 + mhalf;
            #pragma unroll
            for (int r = 0; r < 8; ++r)
                Hs[(mb + r) * Hd + col] = fast_tanhf(c[r] + b1s[col]);
        }
        __syncthreads();

        // (2) Ms = Zs + Hs(32x128) @ W2(128x64) + b2 ; pad rows -> 0
        //     8 tiles (2x4), 1 per wave, K-loop 128/4
        {
            int mi = wv >> 2, ni = wv & 3;
            int row = mi * 16 + ln;
            int col = ni * 16 + ln;
            v8f c = {};
            #pragma unroll
            for (int kk = 0; kk < 32; ++kk) {
                int kb = kk * 4 + khalf;
                v2f a, bb;
                a.x  = Hs[row * Hd + kb];       a.y  = Hs[row * Hd + kb + 1];
                bb.x = W2s[kb * Ld + col];      bb.y = W2s[(kb + 1) * Ld + col];
                c = wmma_f32_4(a, bb, c);
            }
            int mb = mi * 16 + mhalf;
            #pragma unroll
            for (int r = 0; r < 8; ++r) {
                int m = mb + r;
                float v = (m < Sd) ? (Zs[m * Ld + col] + c[r] + b2s[col]) : 0.0f;
                Ms[m * Ld + col] = v;
            }
        }
        __syncthreads();

        // (3) m_p = mean over samples
        if (tid < Ld) {
            float s = 0.0f;
            #pragma unroll
            for (int ss = 0; ss < Sd; ++ss) s += Ms[ss * Ld + tid];
            mpS[tid] = s * (1.0f / (float)Sd);
        }
        __syncthreads();

        // (4) P_p = QI + (M^T M)/S - mp mp^T   (16 tiles (4x4), 2 per wave, K=32)
        for (int q = 0; q < 2; ++q) {
            int tile = wv * 2 + q;
            int mi = tile >> 2, ni = tile & 3;
            int i0 = mi * 16 + ln;
            int j0 = ni * 16 + ln;
            v8f c = {};
            #pragma unroll
            for (int kk = 0; kk < 8; ++kk) {
                int kb = kk * 4 + khalf;
                v2f a, bb;
                a.x  = Ms[kb * Ld + i0];        a.y  = Ms[(kb + 1) * Ld + i0];
                bb.x = Ms[kb * Ld + j0];        bb.y = Ms[(kb + 1) * Ld + j0];
                c = wmma_f32_4(a, bb, c);
            }
            int ib = mi * 16 + mhalf;
            #pragma unroll
            for (int r = 0; r < 8; ++r) {
                int i = ib + r;
                float v = c[r] * (1.0f / (float)Sd) - mpS[i] * mpS[j0];
                if (i == j0) v += Qd[i];
                Ps[i * Ld + j0] = v;
            }
        }
        __syncthreads();

        // (5) diagonal read (BEFORE Cholesky), filter update, outputs
        if (tid < Ld) {
            const int l = tid;
            float dl = Ps[l * Ld + l];
            float kt = kin[(b * Td + t) * Ld + l];
            float Kt = Kin[(b * Td + t) * Ld + l];
            float hf = mpS[l] / dl + kt;
            float Jf = 1.0f / dl + Kt;
            float Pf = 1.0f / Jf;
            float mf = Pf * hf;
            mfS[l]  = mf;
            sPfS[l] = sqrtf(Pf);
            mfO[(b * Td + t) * Ld + l] = mf;
            mpO[(b * Td + t) * Ld + l] = mpS[l];
            pfO[(b * Td + t) * Ld + l] = Pf;
        }
        __syncthreads();

        // (6) z = m_f + sqrt(P_f) * w_t  -> Zs (next step) + z_f output
        for (int p = tid; p < Sd * Ld; p += 256) {
            int s = p >> 6, l = p & 63;
            float zv = mfS[l] + sPfS[l] * win[((t * Sd + s) * Bd + b) * Ld + l];
            Zs[s * Ld + l] = zv;
            zf[s * (Bd * Td * Ld) + b * (Td * Ld) + t * Ld + l] = zv;
        }

        // (6b) prefetch next step's w / k / K rows into cache while the
        //      Cholesky below occupies the critical path (global_prefetch_b8)
        if (t + 1 < Td) {
            if (tid < Sd)
                __builtin_prefetch(&win[(((t + 1) * Sd + tid) * Bd + b) * Ld], 0, 3);
            if (tid == Sd) __builtin_prefetch(&kin[(b * Td + t + 1) * Ld], 0, 3);
            if (tid == Sd + 1) __builtin_prefetch(&Kin[(b * Td + t + 1) * Ld], 0, 3);
        }

        // (7) Cholesky of Ps in place (right-looking, symmetric trailing
        //     update over the full square keeps both triangles valid; 2D
        //     16x16 thread tiling -> no div/mod in the inner loop)
        for (int j = 0; j < Ld; ++j) {
            float pj = Ps[j * Ld + j];          // Ps last written before a barrier
            __syncthreads();                    // everyone read pj before writes
            float inv = 1.0f / sqrtf(pj);
            for (int i = j + tid; i < Ld; i += 256)
                Ps[i * Ld + j] *= inv;          // diag becomes sqrt(pj)
            __syncthreads();
            for (int i = j + 1 + ty; i < Ld; i += 16)
                for (int l = j + 1 + tx; l < Ld; l += 16)
                    Ps[i * Ld + l] -= Ps[i * Ld + j] * Ps[l * Ld + j];
            __syncthreads();
        }
        for (int p = tid; p < Ld * Ld; p += 256) {
            int i = p >> 6, j = p & 63;
            chO[((b * Td + t) * Ld + i) * Ld + j] = (j <= i) ? Ps[p] : 0.0f;
        }
    }
}

extern "C" void kernel_launch(void* const* d_in, const int* in_sizes, int n_in,
                              void* d_out, int out_size, void* d_ws, size_t ws_size,
                              hipStream_t stream) {
    (void)in_sizes; (void)n_in; (void)out_size; (void)d_ws; (void)ws_size;
    const float* kin   = (const float*)d_in[0];
    const float* Kin   = (const float*)d_in[1];
    const float* win   = (const float*)d_in[2];
    const float* m0    = (const float*)d_in[3];
    const float* logQ0 = (const float*)d_in[4];
    const float* logQ  = (const float*)d_in[5];
    const float* W1    = (const float*)d_in[6];
    const float* b1    = (const float*)d_in[7];
    const float* W2    = (const float*)d_in[8];
    const float* b2    = (const float*)d_in[9];
    float* out = (float*)d_out;

    nlf_kernel<<<dim3(Bd), dim3(256), 0, stream>>>(kin, Kin, win, m0, logQ0, logQ,
                                                   W1, b1, W2, b2, out);
}